// DanceRevolution_7567732376225
// MI455X (gfx1250) — compile-verified
//
#include <hip/hip_runtime.h>
#include <hip/hip_bf16.h>
#include <math.h>

#define Bz    4
#define Sz    1800
#define DAz   438
#define DPz   72
#define DMz   200
#define NHz   8
#define DKz   64
#define DVz   64
#define DIz   1024
#define DEz   200
#define NLz   2
#define WINz  100
#define CONDz 10

typedef __attribute__((ext_vector_type(2))) float v2f;
typedef __attribute__((ext_vector_type(4))) float v4f;
typedef __attribute__((ext_vector_type(8))) float v8f;

__device__ __forceinline__ v8f wmma_f32_4(v2f a, v2f b, v8f c) {
  // D = A(16x4 f32) * B(4x16 f32) + C(16x16 f32)
  return __builtin_amdgcn_wmma_f32_16x16x4_f32(false, a, false, b, (short)0, c,
                                               false, false);
}

#define LDSPITCH 36   // 32 + 4 pad: 16B-aligned rows, conflict-free fragment reads

// ---------------------------------------------------------------------------
// GEMM main loop, compile-time K,N: block = 256 threads, C tile 32(M) x 64(N).
// Wave w: msub = w>>2 (0..1), nsub = w&3 (0..3) -> one 16x16 WMMA tile.
// Full k-stages are branch-free (b128 staging when K%4==0); only the final
// partial stage (if K%32 != 0) carries guards/clamps.
// ---------------------------------------------------------------------------
template<int K, int N>
__device__ __forceinline__ v8f gemm_core_t(const float* __restrict__ A,
                                           const float* __restrict__ W,
                                           int mt, int nt, float* Alds)
{
  const int tid = threadIdx.x;
  const int wid = tid >> 5, lane = tid & 31, half = lane >> 4, l15 = lane & 15;
  const int msub = wid >> 2, nsub = wid & 3;
  int bc = nt + nsub * 16 + l15; if (bc > N - 1) bc = N - 1;
  const float* arow = Alds + (size_t)(msub * 16 + l15) * LDSPITCH;
  const float* wp   = W + bc;

  constexpr int KFULL = K & ~31;
  v8f c = {};

  for (int k0 = 0; k0 < KFULL; k0 += 32) {
    __syncthreads();
    if constexpr ((K & 3) == 0) {
      // one b128 per thread covers the whole 32x32 stage
      int row = tid >> 3, c4 = (tid & 7) * 4;
      *(v4f*)(Alds + row * LDSPITCH + c4) =
          *(const v4f*)(A + (size_t)(mt + row) * K + k0 + c4);
    } else {
      #pragma unroll
      for (int pass = 0; pass < 2; ++pass) {
        int e = tid + pass * 256;
        int row = e >> 4, cp = (e & 15) * 2;
        *(v2f*)(Alds + (size_t)row * LDSPITCH + cp) =
            *(const v2f*)(A + (size_t)(mt + row) * K + k0 + cp);
      }
    }
    __syncthreads();
    #pragma unroll
    for (int kk = 0; kk < 8; ++kk) {
      int ka = kk * 4 + half * 2;
      v2f a = *(const v2f*)(arow + ka);
      v2f b; b.x = wp[(size_t)(k0 + ka) * N];
             b.y = wp[(size_t)(k0 + ka + 1) * N];
      c = wmma_f32_4(a, b, c);
    }
  }

  if constexpr (KFULL < K) {
    // tail stage: zero-filled staging, clamped B rows
    __syncthreads();
    #pragma unroll
    for (int pass = 0; pass < 2; ++pass) {
      int e = tid + pass * 256;
      int row = e >> 4, cp = (e & 15) * 2;
      int gk = KFULL + cp;
      v2f v; v.x = 0.f; v.y = 0.f;
      if (gk < K)  // K even, gk even => gk+1 < K too
        v = *(const v2f*)(A + (size_t)(mt + row) * K + gk);
      *(v2f*)(Alds + (size_t)row * LDSPITCH + cp) = v;
    }
    __syncthreads();
    #pragma unroll
    for (int kk = 0; kk < 8; ++kk) {
      int ka = kk * 4 + half * 2;
      v2f a = *(const v2f*)(arow + ka);
      int kr0 = KFULL + ka;     if (kr0 > K - 1) kr0 = K - 1;
      int kr1 = KFULL + ka + 1; if (kr1 > K - 1) kr1 = K - 1;
      v2f b; b.x = wp[(size_t)kr0 * N]; b.y = wp[(size_t)kr1 * N];
      c = wmma_f32_4(a, b, c);
    }
  }
  return c;
}

// ---------------------------------------------------------------------------
// Generic WMMA GEMM: C[M,N] = A[M,K] @ W[K,N] + bias (+res) (+relu)
// ---------------------------------------------------------------------------
template<int K, int N>
__global__ __launch_bounds__(256)
void gemm16_t(const float* __restrict__ A, const float* __restrict__ W,
              const float* __restrict__ bias, const float* __restrict__ res,
              float* __restrict__ C, int M, int relu)
{
  __shared__ float Alds[32 * LDSPITCH];
  const int nt = blockIdx.x * 64, mt = blockIdx.y * 32;
  const int tid = threadIdx.x;
  const int wid = tid >> 5, lane = tid & 31, half = lane >> 4, l15 = lane & 15;
  const int msub = wid >> 2, nsub = wid & 3;

  v8f c = gemm_core_t<K, N>(A, W, mt, nt, Alds);

  #pragma unroll
  for (int r = 0; r < 8; ++r) {
    int row = mt + msub * 16 + r + half * 8;
    int col = nt + nsub * 16 + l15;
    if (row < M && col < N) {
      float v = c[r] + bias[col];
      if (res)  v += res[(size_t)row * N + col];
      if (relu) v  = fmaxf(v, 0.f);
      C[(size_t)row * N + col] = v;
    }
  }
}

// ---------------------------------------------------------------------------
// Embedding GEMM + sinusoid positional encoding (float64 table semantics)
// ---------------------------------------------------------------------------
__global__ __launch_bounds__(256)
void embed16(const float* __restrict__ A, const float* __restrict__ W,
             const float* __restrict__ bias, float* __restrict__ C)
{
  __shared__ float Alds[32 * LDSPITCH];
  const int M = Bz * Sz, N = DMz;
  const int nt = blockIdx.x * 64, mt = blockIdx.y * 32;
  const int tid = threadIdx.x;
  const int wid = tid >> 5, lane = tid & 31, half = lane >> 4, l15 = lane & 15;
  const int msub = wid >> 2, nsub = wid & 3;

  v8f c = gemm_core_t<DAz, DMz>(A, W, mt, nt, Alds);

  #pragma unroll
  for (int r = 0; r < 8; ++r) {
    int row = mt + msub * 16 + r + half * 8;
    int col = nt + nsub * 16 + l15;
    if (row < M && col < N) {
      int s  = row % Sz;
      int i2 = (col >> 1) << 1;
      double freq = pow(10000.0, -(double)i2 / (double)DMz);
      double ang  = (double)(s + 1) * freq;
      float pv = (col & 1) ? (float)cos(ang) : (float)sin(ang);
      C[(size_t)row * N + col] = c[r] + bias[col] + pv;
    }
  }
}

// ---------------------------------------------------------------------------
// Banded (|i-j|<=100) multi-head attention, online softmax, one wave per
// (batch, head, 16-row query tile).
// ---------------------------------------------------------------------------
__global__ __launch_bounds__(32)
void attn_band(const float* __restrict__ Q, const float* __restrict__ Km,
               const float* __restrict__ V, float* __restrict__ O)
{
  const int QT = (Sz + 15) / 16;
  int bid = blockIdx.x;
  int qt = bid % QT; int h = (bid / QT) % NHz; int b = bid / (QT * NHz);
  int qbase = qt * 16;
  int lane = threadIdx.x, half = lane >> 4, l15 = lane & 15;

  int qrow = qbase + l15; if (qrow > Sz - 1) qrow = Sz - 1;
  const float* qp = Q + ((size_t)b * Sz + qrow) * (NHz * DKz) + h * DKz;
  v2f aq[16];
  #pragma unroll
  for (int kk = 0; kk < 16; ++kk) {
    int k0 = kk * 4 + half * 2;
    aq[kk] = *(const v2f*)(qp + k0);          // k0 even, 8B aligned
  }

  float mrow[8], lrow[8];
  #pragma unroll
  for (int r = 0; r < 8; ++r) { mrow[r] = -3.0e38f; lrow[r] = 0.f; }
  v8f acc[4] = {};
  __shared__ float plds[16][17];

  int jt0 = (qbase > WINz) ? (qbase - WINz) / 16 : 0;
  int jt1 = qbase + 15 + WINz; if (jt1 > Sz - 1) jt1 = Sz - 1; jt1 /= 16;

  for (int jt = jt0; jt <= jt1; ++jt) {
    int jbase = jt * 16;
    int krow = jbase + l15; if (krow > Sz - 1) krow = Sz - 1;
    const float* kp = Km + ((size_t)b * Sz + krow) * (NHz * DKz) + h * DKz;
    v8f sc = {};
    #pragma unroll
    for (int kk = 0; kk < 16; ++kk) {
      int k0 = kk * 4 + half * 2;
      v2f bk = *(const v2f*)(kp + k0);
      sc = wmma_f32_4(aq[kk], bk, sc);
    }
    float corr[8];
    #pragma unroll
    for (int r = 0; r < 8; ++r) {
      int row = qbase + r + half * 8;
      int col = jbase + l15;
      int diff = row - col; if (diff < 0) diff = -diff;
      float v = sc[r] * 0.125f;                 // 1/sqrt(64)
      bool ok = (col < Sz) && (diff <= WINz);
      v = ok ? v : -1.0e30f;
      float mx = v;
      mx = fmaxf(mx, __shfl_xor(mx, 1, 16));
      mx = fmaxf(mx, __shfl_xor(mx, 2, 16));
      mx = fmaxf(mx, __shfl_xor(mx, 4, 16));
      mx = fmaxf(mx, __shfl_xor(mx, 8, 16));
      float mn = fmaxf(mrow[r], mx);
      float p  = expf(v - mn);
      float co = expf(mrow[r] - mn);
      float rs = p;
      rs += __shfl_xor(rs, 1, 16);
      rs += __shfl_xor(rs, 2, 16);
      rs += __shfl_xor(rs, 4, 16);
      rs += __shfl_xor(rs, 8, 16);
      lrow[r] = lrow[r] * co + rs;
      mrow[r] = mn;
      corr[r] = co;
      sc[r] = p;
    }
    #pragma unroll
    for (int d = 0; d < 4; ++d)
      #pragma unroll
      for (int r = 0; r < 8; ++r) acc[d][r] *= corr[r];

    __syncthreads();
    #pragma unroll
    for (int r = 0; r < 8; ++r) plds[r + half * 8][l15] = sc[r];
    __syncthreads();

    #pragma unroll
    for (int kk = 0; kk < 4; ++kk) {
      int k0 = kk * 4 + half * 2;
      v2f ap; ap.x = plds[l15][k0]; ap.y = plds[l15][k0 + 1];
      int vr0 = jbase + k0;     if (vr0 > Sz - 1) vr0 = Sz - 1;
      int vr1 = jbase + k0 + 1; if (vr1 > Sz - 1) vr1 = Sz - 1;
      const float* vp0 = V + ((size_t)b * Sz + vr0) * (NHz * DVz) + h * DVz;
      const float* vp1 = V + ((size_t)b * Sz + vr1) * (NHz * DVz) + h * DVz;
      #pragma unroll
      for (int d = 0; d < 4; ++d) {
        v2f bvv; bvv.x = vp0[d * 16 + l15]; bvv.y = vp1[d * 16 + l15];
        acc[d] = wmma_f32_4(ap, bvv, acc[d]);
      }
    }
  }
  #pragma unroll
  for (int r = 0; r < 8; ++r) {
    int row = qbase + r + half * 8;
    if (row < Sz) {
      float inv = 1.f / lrow[r];
      #pragma unroll
      for (int d = 0; d < 4; ++d)
        O[((size_t)b * Sz + row) * (NHz * DVz) + h * DVz + d * 16 + l15] =
            acc[d][r] * inv;
    }
  }
}

// ---------------------------------------------------------------------------
// Row LayerNorm over DMz=200, one wave per row.
// ---------------------------------------------------------------------------
__global__ __launch_bounds__(32)
void ln200(const float* __restrict__ in, const float* __restrict__ g,
           const float* __restrict__ b, float* __restrict__ out)
{
  int row = blockIdx.x; int lane = threadIdx.x;
  const float* p = in + (size_t)row * DMz;
  float vals[7]; float sum = 0.f; int cnt = 0;
  for (int j = lane; j < DMz; j += 32) { float v = p[j]; vals[cnt++] = v; sum += v; }
  for (int m = 16; m; m >>= 1) sum += __shfl_xor(sum, m, 32);
  float mean = sum * (1.f / DMz);
  float vs = 0.f;
  for (int i = 0; i < cnt; ++i) { float d = vals[i] - mean; vs += d * d; }
  for (int m = 16; m; m >>= 1) vs += __shfl_xor(vs, m, 32);
  float rstd = rsqrtf(vs * (1.f / DMz) + 1e-5f);
  int i = 0;
  for (int j = lane; j < DMz; j += 32) {
    out[(size_t)row * DMz + j] = (vals[i] - mean) * rstd * g[j] + b[j];
    ++i;
  }
}

// ---------------------------------------------------------------------------
// Decoder init: copy h/c init, bos -> out_prev, reset grid barrier.
// ---------------------------------------------------------------------------
__global__ __launch_bounds__(256)
void dec_init(const float* __restrict__ h0, const float* __restrict__ c0,
              const float* __restrict__ bos, float* __restrict__ hbuf,
              float* __restrict__ cbuf, float* __restrict__ oprev,
              unsigned* __restrict__ bar)
{
  int i = blockIdx.x * 256 + threadIdx.x;
  if (i < 3 * Bz * DIz) { hbuf[i] = h0[i]; cbuf[i] = c0[i]; }
  if (i < Bz * DPz) oprev[i] = bos[i];
  if (i == 0) *bar = 0u;
}

// ---------------------------------------------------------------------------
// Persistent autoregressive 3-layer LSTM decoder. 96 WGs x 256 threads.
//   wg/32 = cell (0..2); (wg%32)*32 = 32 hidden units owned.
//   8 waves per WG each compute one 16-col gate tile via WMMA (M=4 padded 16).
//   h double-buffered on step parity; 2 grid barriers per step.
//   All LSTM weights (~87 MB fp32) stay resident in the 192 MB L2.
// ---------------------------------------------------------------------------
#define DEC_NWG 96
#define DEC_NT  256

__global__ __launch_bounds__(DEC_NT)
void decoder_persist(
    const float* __restrict__ tgt_seq, const int* __restrict__ epoch_p,
    const float* __restrict__ tgtW, const float* __restrict__ tgtB,
    const float* __restrict__ Wih1, const float* __restrict__ Whh1,
    const float* __restrict__ bih1, const float* __restrict__ bhh1,
    const float* __restrict__ Wih2, const float* __restrict__ Whh2,
    const float* __restrict__ bih2, const float* __restrict__ bhh2,
    const float* __restrict__ Wih3, const float* __restrict__ Whh3,
    const float* __restrict__ bih3, const float* __restrict__ bhh3,
    const float* __restrict__ outW, const float* __restrict__ outB,
    const float* __restrict__ enc,
    float* __restrict__ hbuf, float* __restrict__ cbuf,
    float* __restrict__ oprev, unsigned* __restrict__ bar,
    float* __restrict__ dout)
{
  __shared__ float e_lds[4][DEz];
  __shared__ float glds[4][4][32];
  __shared__ float red[8];

  const int tid  = threadIdx.x;
  const int wid  = tid >> 5, lane = tid & 31, half = lane >> 4, l15 = lane & 15;
  const int wg   = blockIdx.x;
  const int cell = wg / 32;
  const int ubase = (wg % 32) * 32;
  const int gate  = wid >> 1;
  const int gcol  = gate * 1024 + ubase + (wid & 1) * 16 + l15;
  const int mA    = (l15 < 4) ? l15 : 3;   // clamp padded rows

  const float* Wih = (cell == 0) ? Wih1 : (cell == 1) ? Wih2 : Wih3;
  const float* Whh = (cell == 0) ? Whh1 : (cell == 1) ? Whh2 : Whh3;
  const float* bih = (cell == 0) ? bih1 : (cell == 1) ? bih2 : bih3;
  const float* bhh = (cell == 0) ? bhh1 : (cell == 1) ? bhh2 : bhh3;
  const float* wihp = Wih + gcol;          // column base, constant offsets below
  const float* whhp = Whh + gcol;

  const int p   = (int)((double)epoch_p[0] * 0.01);  // matches int(epoch*0.01)
  const int per = p + CONDz;
  unsigned bcount = 0;
  const unsigned nwg = gridDim.x;

  for (int t = 0; t < Sz; ++t) {
    const int pold = t & 1, pnew = pold ^ 1;

    // ---- e = dec_in @ tgtW + tgtB (cell-0 blocks compute it redundantly) ----
    if (cell == 0) {
      bool use_gt = (per > 0) ? ((t % per) >= p) : true;
      for (int idx = tid; idx < 4 * DEz; idx += DEC_NT) {
        int m = idx / DEz, dc = idx % DEz;
        float s = tgtB[dc];
        for (int j = 0; j < DPz; ++j) {
          float di = use_gt ? tgt_seq[((size_t)m * Sz + t) * DPz + j]
                            : oprev[m * DPz + j];
          s += di * tgtW[j * DEz + dc];
        }
        e_lds[m][dc] = s;
      }
      __syncthreads();
    }

    // ---- gate GEMM: x @ Wih  +  h_old @ Whh ----
    v8f c = {};
    if (cell == 0) {
      #pragma unroll 8
      for (int k = 0; k < DEz; k += 4) {
        int ka = k + half * 2;                    // even -> 8B aligned
        v2f a  = *(const v2f*)(&e_lds[mA][ka]);
        v2f bm; bm.x = wihp[(size_t)ka * 4096];
                bm.y = wihp[(size_t)(ka + 1) * 4096];
        c = wmma_f32_4(a, bm, c);
      }
    } else {
      const float* x = hbuf + ((size_t)(pold * 3 + (cell - 1)) * 4) * DIz;
      #pragma unroll 8
      for (int k = 0; k < DIz; k += 4) {
        int ka = k + half * 2;
        v2f a  = *(const v2f*)(x + mA * DIz + ka);
        v2f bm; bm.x = wihp[(size_t)ka * 4096];
                bm.y = wihp[(size_t)(ka + 1) * 4096];
        c = wmma_f32_4(a, bm, c);
      }
    }
    {
      const float* hv = hbuf + ((size_t)(pold * 3 + cell) * 4) * DIz;
      #pragma unroll 8
      for (int k = 0; k < DIz; k += 4) {
        int ka = k + half * 2;
        v2f a  = *(const v2f*)(hv + mA * DIz + ka);
        v2f bm; bm.x = whhp[(size_t)ka * 4096];
                bm.y = whhp[(size_t)(ka + 1) * 4096];
        c = wmma_f32_4(a, bm, c);
      }
    }
    if (half == 0) {
      float bsum = bih[gcol] + bhh[gcol];
      #pragma unroll
      for (int r = 0; r < 4; ++r)
        glds[gate][r][(wid & 1) * 16 + l15] = c[r] + bsum;
    }
    __syncthreads();

    // ---- LSTM cell-state update for the 32 owned units x 4 batch ----
    if (tid < 128) {
      int m = tid & 3, ul = tid >> 2;
      int u = ubase + ul;
      float iv = glds[0][m][ul], fv = glds[1][m][ul];
      float gv = glds[2][m][ul], ov = glds[3][m][ul];
      iv = 1.f / (1.f + expf(-iv));
      fv = 1.f / (1.f + expf(-fv));
      gv = tanhf(gv);
      ov = 1.f / (1.f + expf(-ov));
      size_t ci = ((size_t)cell * 4 + m) * DIz + u;
      float cn = fv * cbuf[ci] + iv * gv;
      cbuf[ci] = cn;
      hbuf[((size_t)(pnew * 3 + cell) * 4 + m) * DIz + u] = ov * tanhf(cn);
    }
    __syncthreads();

    // ---- grid barrier #1: all new h written ----
    bcount++;
    __threadfence();
    if (tid == 0) {
      __hip_atomic_fetch_add(bar, 1u, __ATOMIC_ACQ_REL, __HIP_MEMORY_SCOPE_AGENT);
      unsigned tv = bcount * nwg;
      while (__hip_atomic_load(bar, __ATOMIC_ACQUIRE, __HIP_MEMORY_SCOPE_AGENT) < tv)
        __builtin_amdgcn_s_sleep(2);
    }
    __syncthreads();

    // ---- out[t] = [h2 || enc_t] @ outW + outB, one column per WG (0..71) ----
    // 8 waves: m = wid&3 (batch), seg = wid>>2 (K halves); wave-level shfl
    // reduction, single block barrier, 4-thread finish.
    if (wg < DPz) {
      const int col = wg;
      const int m   = wid & 3;
      const int seg = wid >> 2;           // 0..1, each 612 of K=1224
      const float* h2 = hbuf + ((size_t)(pnew * 3 + 2) * 4 + m) * DIz;
      const float* ep = enc + ((size_t)m * Sz + t) * DMz;
      float part = 0.f;
      for (int k = seg * 612 + lane; k < (seg + 1) * 612; k += 32) {
        float a = (k < DIz) ? h2[k] : ep[k - DIz];
        part += a * outW[(size_t)k * DPz + col];
      }
      part += __shfl_xor(part, 1, 32);
      part += __shfl_xor(part, 2, 32);
      part += __shfl_xor(part, 4, 32);
      part += __shfl_xor(part, 8, 32);
      part += __shfl_xor(part, 16, 32);
      if (lane == 0) red[wid] = part;
      __syncthreads();
      if (tid < 4) {
        float v = red[tid] + red[tid + 4] + outB[col];
        dout[((size_t)tid * Sz + t) * DPz + col] = v;
        oprev[tid * DPz + col] = v;
      }
    }

    // ---- grid barrier #2: out_prev visible for next step's dec_in ----
    bcount++;
    __threadfence();
    if (tid == 0) {
      __hip_atomic_fetch_add(bar, 1u, __ATOMIC_ACQ_REL, __HIP_MEMORY_SCOPE_AGENT);
      unsigned tv = bcount * nwg;
      while (__hip_atomic_load(bar, __ATOMIC_ACQUIRE, __HIP_MEMORY_SCOPE_AGENT) < tv)
        __builtin_amdgcn_s_sleep(2);
    }
    __syncthreads();
  }
}

// ---------------------------------------------------------------------------
extern "C" void kernel_launch(void* const* d_in, const int* in_sizes, int n_in,
                              void* d_out, int out_size, void* d_ws, size_t ws_size,
                              hipStream_t stream)
{
  const float* src   = (const float*)d_in[0];
  const float* tgt   = (const float*)d_in[1];
  const float* bos   = (const float*)d_in[2];
  const float* hinit = (const float*)d_in[3];
  const float* cinit = (const float*)d_in[4];
  const int*   epoch = (const int*)  d_in[5];
  const float* eW    = (const float*)d_in[6];
  const float* eB    = (const float*)d_in[7];
  const float* Wq    = (const float*)d_in[8];
  const float* bq    = (const float*)d_in[9];
  const float* Wk    = (const float*)d_in[10];
  const float* bk    = (const float*)d_in[11];
  const float* Wv    = (const float*)d_in[12];
  const float* bv    = (const float*)d_in[13];
  const float* Wo    = (const float*)d_in[14];
  const float* bo    = (const float*)d_in[15];
  const float* ln1g  = (const float*)d_in[16];
  const float* ln1b  = (const float*)d_in[17];
  const float* W1    = (const float*)d_in[18];
  const float* b1    = (const float*)d_in[19];
  const float* W2    = (const float*)d_in[20];
  const float* b2    = (const float*)d_in[21];
  const float* ln2g  = (const float*)d_in[22];
  const float* ln2b  = (const float*)d_in[23];
  const float* tgtW  = (const float*)d_in[24];
  const float* tgtB  = (const float*)d_in[25];
  const float* Wih1  = (const float*)d_in[26];
  const float* Whh1  = (const float*)d_in[27];
  const float* bih1  = (const float*)d_in[28];
  const float* bhh1  = (const float*)d_in[29];
  const float* Wih2  = (const float*)d_in[30];
  const float* Whh2  = (const float*)d_in[31];
  const float* bih2  = (const float*)d_in[32];
  const float* bhh2  = (const float*)d_in[33];
  const float* Wih3  = (const float*)d_in[34];
  const float* Whh3  = (const float*)d_in[35];
  const float* bih3  = (const float*)d_in[36];
  const float* bhh3  = (const float*)d_in[37];
  const float* outW  = (const float*)d_in[38];
  const float* outB  = (const float*)d_in[39];
  float* dout = (float*)d_out;

  const int M = Bz * Sz;                 // 7200, multiple of 32
  float* ws = (float*)d_ws;
  size_t off = 0;
  float* X    = ws + off; off += (size_t)M * DMz;   // activations
  float* Qb   = ws + off; off += (size_t)M * 512;   // Q (aliased by FFN later)
  float* Kb   = ws + off; off += (size_t)M * 512;   // K (aliased by FFN later)
  float* Vb   = ws + off; off += (size_t)M * 512;   // V
  float* AO   = ws + off; off += (size_t)M * 512;   // attention out
  float* G1   = ws + off; off += (size_t)M * DMz;   // pre-LN buffer
  float* FF   = Qb;                                  // [M,1024] aliases Q|K
  float* hbuf = ws + off; off += 2 * 3 * Bz * DIz;
  float* cbuf = ws + off; off += 3 * Bz * DIz;
  float* oprv = ws + off; off += Bz * DPz;
  unsigned* bar = (unsigned*)(ws + off); off += 16;

  dim3 blk(256);
  const int QT = (Sz + 15) / 16;

  embed16<<<dim3((DMz + 63) / 64, M / 32), blk, 0, stream>>>(src, eW, eB, X);

  for (int l = 0; l < NLz; ++l) {
    gemm16_t<DMz, 512><<<dim3(512 / 64, M / 32), blk, 0, stream>>>(
        X, Wq + (size_t)l * DMz * 512, bq + l * 512, nullptr, Qb, M, 0);
    gemm16_t<DMz, 512><<<dim3(512 / 64, M / 32), blk, 0, stream>>>(
        X, Wk + (size_t)l * DMz * 512, bk + l * 512, nullptr, Kb, M, 0);
    gemm16_t<DMz, 512><<<dim3(512 / 64, M / 32), blk, 0, stream>>>(
        X, Wv + (size_t)l * DMz * 512, bv + l * 512, nullptr, Vb, M, 0);
    attn_band<<<Bz * NHz * QT, dim3(32), 0, stream>>>(Qb, Kb, Vb, AO);
    gemm16_t<512, DMz><<<dim3((DMz + 63) / 64, M / 32), blk, 0, stream>>>(
        AO, Wo + (size_t)l * 512 * DMz, bo + l * DMz, X, G1, M, 0);
    ln200<<<M, dim3(32), 0, stream>>>(G1, ln1g + l * DMz, ln1b + l * DMz, X);
    gemm16_t<DMz, DIz><<<dim3(DIz / 64, M / 32), blk, 0, stream>>>(
        X, W1 + (size_t)l * DMz * DIz, b1 + l * DIz, nullptr, FF, M, 1);
    gemm16_t<DIz, DMz><<<dim3((DMz + 63) / 64, M / 32), blk, 0, stream>>>(
        FF, W2 + (size_t)l * DIz * DMz, b2 + l * DMz, X, G1, M, 0);
    ln200<<<M, dim3(32), 0, stream>>>(G1, ln2g + l * DMz, ln2b + l * DMz, X);
  }

  dec_init<<<48, 256, 0, stream>>>(hinit, cinit, bos, hbuf, cbuf, oprv, bar);
  decoder_persist<<<DEC_NWG, DEC_NT, 0, stream>>>(
      tgt, epoch, tgtW, tgtB,
      Wih1, Whh1, bih1, bhh1, Wih2, Whh2, bih2, bhh2, Wih3, Whh3, bih3, bhh3,
      outW, outB, X, hbuf, cbuf, oprv, bar, dout);

  (void)in_sizes; (void)n_in; (void)out_size; (void)ws_size;
}